// PyG_PrototypicalNetwork_16947940950132
// MI455X (gfx1250) — compile-verified
//
#include <hip/hip_runtime.h>
#include <hip/hip_bf16.h>

// MI455X / gfx1250 (CDNA5, wave32). Mahalanobis prototypical-network head.
//
// Pipeline (all on `stream`, graph-capture safe):
//   1) stats_kernel      : class counts, class means, task mean
//   2) build_m_kernel    : M_c = lam*cov_c + (1-lam)*task_cov + 0.1*I   (2x 512x512, LDS-tiled)
//   3) power_iter_kernel : lambda_max(M_c) -> alpha_c = 2/(lmin+1.10*lmax)
//   4) init_x_kernel     : X0 = alpha*I
//   5) 12x Newton-Schulz : R = 2I - M X ; X = X R      (fp32 WMMA GEMMs -> P = M^-1)
//   6) mahal_kernel      : logits[q,c] = -(mu_c - q)^T P_c (mu_c - q)   (fused WMMA GEMM + reduce)

typedef float v2f __attribute__((ext_vector_type(2)));
typedef float v8f __attribute__((ext_vector_type(8)));

#define D_FEAT 512
#define NS     512
#define NQ     16384
#define NCLS   2
#define TM     64      // queries per block in mahal_kernel
#define QSTRIDE 516    // LDS row stride (516%64==4 -> conflict-free 16-row frag reads)

// ---------------------------------------------------------------------------
// 1) per-feature stats: counts, per-class means, overall mean
__global__ void stats_kernel(const float* __restrict__ X, const int* __restrict__ lab,
                             float* __restrict__ counts, float* __restrict__ mu_all,
                             float* __restrict__ mu_cls) {
  int j = blockIdx.x * blockDim.x + threadIdx.x;   // 0..511 feature dim
  if (j >= D_FEAT) return;
  float sa = 0.f, s0 = 0.f, s1 = 0.f;
  int n0 = 0, n1 = 0;
  for (int s = 0; s < NS; ++s) {
    float x = X[(size_t)s * D_FEAT + j];           // coalesced across j
    int   L = lab[s];                              // broadcast
    sa += x;
    if (L == 0) { s0 += x; ++n0; } else { s1 += x; ++n1; }
  }
  mu_all[j] = sa * (1.f / (float)NS);
  mu_cls[j]          = s0 / (float)n0;
  mu_cls[D_FEAT + j] = s1 / (float)n1;
  if (j == 0) { counts[0] = (float)n0; counts[1] = (float)n1; }
}

// ---------------------------------------------------------------------------
// 2) M_c[i][j] for both classes; LDS-tiled over samples
#define BM_TS 32
__global__ void build_m_kernel(const float* __restrict__ X, const int* __restrict__ lab,
                               const float* __restrict__ counts,
                               const float* __restrict__ mu_all,
                               const float* __restrict__ mu_cls,
                               float* __restrict__ M) {
  __shared__ float xi[BM_TS][17];
  __shared__ float xj[BM_TS][17];
  __shared__ int   lbs[BM_TS];
  int ti = threadIdx.x, tj = threadIdx.y;          // 16x16 block
  int i0 = blockIdx.x * 16, j0 = blockIdx.y * 16;
  int i = i0 + ti, j = j0 + tj;
  float mai = mu_all[i], maj = mu_all[j];
  float m0i = mu_cls[i], m0j = mu_cls[j];
  float m1i = mu_cls[D_FEAT + i], m1j = mu_cls[D_FEAT + j];
  float sA = 0.f, s0 = 0.f, s1 = 0.f;
  int lin = tj * 16 + ti;
  for (int sb = 0; sb < NS; sb += BM_TS) {
    for (int t = lin; t < BM_TS * 16; t += 256) {
      int ss = t >> 4, cc = t & 15;
      xi[ss][cc] = X[(size_t)(sb + ss) * D_FEAT + i0 + cc];
      xj[ss][cc] = X[(size_t)(sb + ss) * D_FEAT + j0 + cc];
    }
    if (lin < BM_TS) lbs[lin] = lab[sb + lin];
    __syncthreads();
#pragma unroll 4
    for (int s = 0; s < BM_TS; ++s) {
      float vi = xi[s][ti], vj = xj[s][tj];
      int   L  = lbs[s];
      sA += (vi - mai) * (vj - maj);
      float p0 = (vi - m0i) * (vj - m0j);
      float p1 = (vi - m1i) * (vj - m1j);
      s0 += (L == 0) ? p0 : 0.f;
      s1 += (L == 1) ? p1 : 0.f;
    }
    __syncthreads();
  }
  float n0 = counts[0], n1 = counts[1];
  float task = sA * (1.f / (float)(NS - 1));
  float cov0 = s0 / (n0 - 1.f);
  float cov1 = s1 / (n1 - 1.f);
  float lam0 = fminf(n0 / (n0 + 1.f), 0.1f);
  float lam1 = fminf(n1 / (n1 + 1.f), 0.1f);
  float diag = (i == j) ? 0.1f : 0.f;
  M[(size_t)i * D_FEAT + j] = lam0 * cov0 + (1.f - lam0) * task + diag;
  M[(size_t)D_FEAT * D_FEAT + (size_t)i * D_FEAT + j] =
      lam1 * cov1 + (1.f - lam1) * task + diag;
}

// ---------------------------------------------------------------------------
// 3) power iteration: lambda_max estimate -> alpha = 2/(lmin + 1.10*lmax)
__global__ void power_iter_kernel(const float* __restrict__ M, float* __restrict__ alphas) {
  const float* Mc = M + (size_t)blockIdx.x * D_FEAT * D_FEAT;
  __shared__ float v[D_FEAT];
  __shared__ float red[D_FEAT];
  int t = threadIdx.x;                             // 512 threads
  v[t] = 1.0f + 0.001f * (float)(t % 31);
  __syncthreads();
  float lam = 1.f;
  for (int it = 0; it < 32; ++it) {
    // M is symmetric: w_j = sum_i M[i][j] v[i]  (coalesced over j = t)
    float w = 0.f;
    for (int i = 0; i < D_FEAT; ++i) w += Mc[(size_t)i * D_FEAT + t] * v[i];
    red[t] = w * w;
    __syncthreads();
    for (int off = 256; off > 0; off >>= 1) {
      if (t < off) red[t] += red[t + off];
      __syncthreads();
    }
    float nrm = sqrtf(red[0]);
    lam = nrm;
    v[t] = w / nrm;
    __syncthreads();
  }
  if (t == 0) alphas[blockIdx.x] = 2.0f / (0.1f + 1.10f * lam);
}

// ---------------------------------------------------------------------------
// 4) X0 = alpha * I
__global__ void init_x_kernel(const float* __restrict__ alphas, float* __restrict__ Xb) {
  size_t idx = (size_t)blockIdx.x * blockDim.x + threadIdx.x;
  const size_t per = (size_t)D_FEAT * D_FEAT;
  if (idx >= 2 * per) return;
  int    c = (int)(idx / per);
  size_t r = idx % per;
  int i = (int)(r / D_FEAT), j = (int)(r % D_FEAT);
  Xb[idx] = (i == j) ? alphas[c] : 0.f;
}

// ---------------------------------------------------------------------------
// 5) Newton-Schulz GEMM, fp32 WMMA 16x16x4.
//    mode 0: C = 2I - A*B ; mode 1: C = A*B.  grid (32,4,2), block 256.
__global__ void __launch_bounds__(256)
newton_gemm_kernel(const float* __restrict__ A, const float* __restrict__ B,
                   float* __restrict__ C, int mode) {
  const size_t off = (size_t)blockIdx.z * D_FEAT * D_FEAT;
  const float* Ac = A + off;
  const float* Bc = B + off;
  float*       Cc = C + off;
  int lane = threadIdx.x & 31;
  int wave = threadIdx.x >> 5;                     // 0..7
  int half = lane >> 4;                            // half-wave (K +2 / M +8 split)
  int l16  = lane & 15;
  int m0 = blockIdx.x * 16;                        // 32 row tiles
  int n0 = (blockIdx.y * 8 + wave) * 16;           // 32 col tiles
  const float* arow = Ac + (size_t)(m0 + l16) * D_FEAT + 2 * half;  // A[M=l16][K=2h+v]
  const float* bcol = Bc + (size_t)(2 * half) * D_FEAT + n0 + l16;  // B[K=2h+v][N=l16]
  v8f acc = {};
#pragma unroll 4
  for (int k = 0; k < D_FEAT; k += 4) {
    v2f a = *(const v2f*)(arow + k);
    v2f b;
    b.x = bcol[(size_t)k * D_FEAT];
    b.y = bcol[(size_t)(k + 1) * D_FEAT];
    acc = __builtin_amdgcn_wmma_f32_16x16x4_f32(false, a, false, b, (short)0, acc,
                                                false, false);
  }
  int mrow = m0 + 8 * half;
  int ncol = n0 + l16;
#pragma unroll
  for (int r = 0; r < 8; ++r) {
    int   m   = mrow + r;                          // C: vgpr r -> M = m0 + r (+8 hi half)
    float val = acc[r];
    if (mode == 0) val = ((m == ncol) ? 2.0f : 0.0f) - val;
    Cc[(size_t)m * D_FEAT + ncol] = val;
  }
}

// ---------------------------------------------------------------------------
// 6) fused Mahalanobis: out[q,c] = -sum_j (D P)[q,j] * D[q,j],  D = mu_c - q
__global__ void __launch_bounds__(256)
mahal_kernel(const float* __restrict__ Q, const float* __restrict__ mu_cls,
             const float* __restrict__ P, float* __restrict__ out) {
  extern __shared__ float smem[];
  float* Ds   = smem;                      // TM x QSTRIDE : holds mu_c - q (in place)
  float* muS  = Ds + TM * QSTRIDE;         // 2 x 512
  float* accQ = muS + 2 * D_FEAT;          // 2 x TM
  int tid = threadIdx.x;
  int q0  = blockIdx.x * TM;

  for (int t = tid; t < 2 * D_FEAT; t += 256) muS[t] = mu_cls[t];
  for (int t = tid; t < NCLS * TM; t += 256) accQ[t] = 0.f;
  __syncthreads();
  // load queries and convert directly to D0 = mu0 - q (coalesced global reads)
  for (int t = tid; t < TM * D_FEAT; t += 256) {
    int r = t >> 9, col = t & (D_FEAT - 1);
    Ds[r * QSTRIDE + col] = muS[col] - Q[(size_t)(q0 + r) * D_FEAT + col];
  }
  __syncthreads();

  int lane = tid & 31, wave = tid >> 5;
  int half = lane >> 4, l16 = lane & 15;
  int rt = wave & 3;                        // row tile (4 x 16 = 64 rows)
  int cg = wave >> 2;                       // column-group (strided ntiles)
  const float* drow = Ds + (rt * 16 + l16) * QSTRIDE + 2 * half;

  for (int c = 0; c < NCLS; ++c) {
    if (c == 1) {                           // D1 = D0 + (mu1 - mu0), updated in place
      __syncthreads();
      for (int t = tid; t < TM * D_FEAT; t += 256) {
        int r = t >> 9, col = t & (D_FEAT - 1);
        Ds[r * QSTRIDE + col] += muS[D_FEAT + col] - muS[col];
      }
      __syncthreads();
    }
    const float* Pc = P + (size_t)c * D_FEAT * D_FEAT;
    float racc[8];
#pragma unroll
    for (int r = 0; r < 8; ++r) racc[r] = 0.f;

    for (int nt = cg; nt < D_FEAT / 16; nt += 2) {
      int n0 = nt * 16;
      const float* pcol = Pc + (size_t)(2 * half) * D_FEAT + n0 + l16;
      v8f acc = {};
#pragma unroll 4
      for (int k = 0; k < D_FEAT; k += 4) {
        v2f a = *(const v2f*)(drow + k);    // LDS b64, conflict-free (stride 516)
        v2f b;
        b.x = pcol[(size_t)k * D_FEAT];     // L2-resident P, coalesced 16-lane rows
        b.y = pcol[(size_t)(k + 1) * D_FEAT];
        acc = __builtin_amdgcn_wmma_f32_16x16x4_f32(false, a, false, b, (short)0, acc,
                                                    false, false);
      }
      // epilogue: h*d, butterfly-reduce the 16 N-lanes of each half-wave
      int   ncol = n0 + l16;
#pragma unroll
      for (int r = 0; r < 8; ++r) {
        int   m    = rt * 16 + r + 8 * half;
        float dval = Ds[m * QSTRIDE + ncol];
        float p    = acc[r] * dval;
        p += __shfl_xor(p, 1);
        p += __shfl_xor(p, 2);
        p += __shfl_xor(p, 4);
        p += __shfl_xor(p, 8);
        if (l16 == 0) racc[r] += p;
      }
    }
    if (l16 == 0) {
#pragma unroll
      for (int r = 0; r < 8; ++r) {
        int m = rt * 16 + r + 8 * half;
        atomicAdd(&accQ[c * TM + m], racc[r]);  // ds_add_f32
      }
    }
  }
  __syncthreads();
  for (int t = tid; t < NCLS * TM; t += 256) {
    int c = t / TM, m = t % TM;
    out[(size_t)(q0 + m) * NCLS + c] = -accQ[c * TM + m];
  }
}

// ---------------------------------------------------------------------------
extern "C" void kernel_launch(void* const* d_in, const int* in_sizes, int n_in,
                              void* d_out, int out_size, void* d_ws, size_t ws_size,
                              hipStream_t stream) {
  const float* X   = (const float*)d_in[0];   // support_features [512,512]
  const int*   lab = (const int*)d_in[1];     // support_labels   [512]
  const float* Q   = (const float*)d_in[2];   // query_features   [16384,512]
  float*       out = (float*)d_out;           // logits [16384,2]
  float*       ws  = (float*)d_ws;

  const size_t per = (size_t)D_FEAT * D_FEAT;
  float* counts = ws;             // 2
  float* mu_all = ws + 16;        // 512
  float* mu_cls = ws + 528;       // 2*512
  float* alphas = ws + 1552;      // 2
  float* M      = ws + 2048;      // 2*per
  float* Xb0    = M + 2 * per;    // 2*per
  float* Xb1    = Xb0 + 2 * per;  // 2*per
  float* R      = Xb1 + 2 * per;  // 2*per   (total ~8.4 MB of d_ws)

  stats_kernel<<<2, 256, 0, stream>>>(X, lab, counts, mu_all, mu_cls);
  build_m_kernel<<<dim3(32, 32), dim3(16, 16), 0, stream>>>(X, lab, counts, mu_all,
                                                            mu_cls, M);
  power_iter_kernel<<<2, 512, 0, stream>>>(M, alphas);
  init_x_kernel<<<(int)((2 * per + 255) / 256), 256, 0, stream>>>(alphas, Xb0);

  float* Xc = Xb0;
  float* Xn = Xb1;
  for (int it = 0; it < 12; ++it) {        // Newton-Schulz: X <- X(2I - M X)
    newton_gemm_kernel<<<dim3(32, 4, 2), 256, 0, stream>>>(M, Xc, R, 0);
    newton_gemm_kernel<<<dim3(32, 4, 2), 256, 0, stream>>>(Xc, R, Xn, 1);
    float* t = Xc; Xc = Xn; Xn = t;
  }
  // after 12 swaps the converged inverse is in Xc (== Xb0)

  size_t smem = (size_t)(TM * QSTRIDE + 2 * D_FEAT + NCLS * TM) * sizeof(float);
  mahal_kernel<<<NQ / TM, 256, smem, stream>>>(Q, mu_cls, Xc, out);
}